// ConsciousCrossEntropy_64527588655435
// MI455X (gfx1250) — compile-verified
//
#include <hip/hip_runtime.h>

// ---------------------------------------------------------------------------
// ConsciousCrossEntropy on MI455X (gfx1250, wave32)
//
// target:      [256,128,42] f32 one-hot
// predictions: [256,128,42] f32 softmax
// out:         scalar f32
//
// Memory-bound (11 MB in, 4 B out). Per-position 42-length dot products are
// computed with chained V_WMMA_F32_16X16X4_F32: a 16-position tile forms
// C = A(16xK tgt) x B(Kx16 pred^T); diag(C) = prob of true class.
// ---------------------------------------------------------------------------

typedef __attribute__((ext_vector_type(2))) float v2f;
typedef __attribute__((ext_vector_type(8))) float v8f;

#define CE_BATCH 256
#define CE_SEQ 128
#define CE_ALPHA 42
#define CE_IGNORE 40

__global__ __launch_bounds__(256) void ce_row_kernel(
    const float* __restrict__ target,
    const float* __restrict__ pred,
    float* __restrict__ partial /* [CE_BATCH] */) {
  const int b    = blockIdx.x;
  const int tid  = threadIdx.x;
  const int wave = tid >> 5;   // 8 waves, wave w owns positions 16w..16w+15
  const int lane = tid & 31;
  const int half = lane >> 4;  // K sub-group within the lane pair halves
  const int mn   = lane & 15;  // A: row M / B: col N handled by this lane

  const int s0 = wave * 16;
  const size_t rowbase = ((size_t)b * CE_SEQ + (size_t)(s0 + mn)) * CE_ALPHA;
  const float* trow = target + rowbase;
  const float* prow = pred + rowbase;

  // Accumulate C = sum_k A(16x4) * B(4x16); K padded 42 -> 44 with zeros.
  v8f c = {0.f, 0.f, 0.f, 0.f, 0.f, 0.f, 0.f, 0.f};
#pragma unroll
  for (int k0 = 0; k0 < 44; k0 += 4) {
    const int ka = k0 + 2 * half;  // first of the two K elements this lane supplies
    v2f a  = {0.f, 0.f};
    v2f bm = {0.f, 0.f};
    if (ka + 1 < CE_ALPHA) {       // CE_ALPHA even -> pair fully valid or fully padded
      a.x  = trow[ka];
      a.y  = trow[ka + 1];
      bm.x = prow[ka];
      bm.y = prow[ka + 1];
    }
    // 8 args: (neg_a, A, neg_b, B, c_mod, C, reuse_a, reuse_b)
    c = __builtin_amdgcn_wmma_f32_16x16x4_f32(
        /*neg_a=*/false, a, /*neg_b=*/false, bm,
        /*c_mod=*/(short)0, c, /*reuse_a=*/false, /*reuse_b=*/false);
  }

  // Diagonal of C (16x16 f32 layout): VGPR r, lane L -> (M = r + 8*(L>>4), N = L&15).
  // diag n<8  : lane n,      c[n]
  // diag n>=8 : lane n+16,   c[n-8]
  const bool has_diag = (lane < 8) || (lane >= 24);
  const int  didx     = lane & 7;  // correct C component for both active groups
  float prob = 0.f;
#pragma unroll
  for (int r = 0; r < 8; ++r) prob = (r == didx) ? c[r] : prob;

  __shared__ float ls[CE_SEQ];  // mask * (-log prob) per position
  __shared__ float ms[CE_SEQ];  // mask per position
  if (has_diag) {
    const int s = s0 + mn;                       // mn == diag index for active lanes
    const float mask = 1.0f - trow[CE_IGNORE];   // 0 iff true class is the ignore token
    const float loss = -__logf(prob);
    ls[s] = mask * loss;
    ms[s] = mask;
  }
  __syncthreads();

  // Deterministic tree reduction over 128 positions.
#pragma unroll
  for (int stride = CE_SEQ / 2; stride > 0; stride >>= 1) {
    if (tid < stride) {
      ls[tid] += ls[tid + stride];
      ms[tid] += ms[tid + stride];
    }
    __syncthreads();
  }

  if (tid == 0) {
    // Faithful to source: divide by (SEQ - nops) = count of IGNORED positions
    // (may be 0 -> inf, matching the reference's IEEE behavior).
    partial[b] = ls[0] / ((float)CE_SEQ - ms[0]);
  }
}

__global__ __launch_bounds__(CE_BATCH) void ce_final_kernel(
    const float* __restrict__ partial, float* __restrict__ out) {
  __shared__ float sm[CE_BATCH];
  const int tid = threadIdx.x;
  sm[tid] = partial[tid];
  __syncthreads();
#pragma unroll
  for (int stride = CE_BATCH / 2; stride > 0; stride >>= 1) {
    if (tid < stride) sm[tid] += sm[tid + stride];
    __syncthreads();
  }
  if (tid == 0) out[0] = sm[0] * (1.0f / (float)CE_BATCH);
}

extern "C" void kernel_launch(void* const* d_in, const int* in_sizes, int n_in,
                              void* d_out, int out_size, void* d_ws, size_t ws_size,
                              hipStream_t stream) {
  (void)in_sizes; (void)n_in; (void)out_size; (void)ws_size;
  const float* target = (const float*)d_in[0];       // [256,128,42] f32
  const float* pred   = (const float*)d_in[1];       // [256,128,42] f32
  float* partial      = (float*)d_ws;                // 256 floats of scratch
  float* out          = (float*)d_out;               // scalar f32

  ce_row_kernel<<<CE_BATCH, 256, 0, stream>>>(target, pred, partial);
  ce_final_kernel<<<1, CE_BATCH, 0, stream>>>(partial, out);
}